// Self_Attention_43576738185709
// MI455X (gfx1250) — compile-verified
//
#include <hip/hip_runtime.h>
#include <hip/hip_bf16.h>

typedef float v2f __attribute__((ext_vector_type(2)));
typedef float v8f __attribute__((ext_vector_type(8)));

#define BATCH 4
#define SEQ   1024
#define DIM   512
#define NHEAD 8
#define HDK   64   // per-head dk = dv
#define SCALE 0.04419417382415922f   // 1/sqrt(512)
#define SCPAD 1028                   // LDS row stride (floats), 1028%64==4 -> conflict-free

// ---------------------------------------------------------------------------
// LayerNorm over last dim (512), affine. One block per row.
// ---------------------------------------------------------------------------
__global__ __launch_bounds__(256) void ln_kernel(const float* __restrict__ x,
                                                 const float* __restrict__ g,
                                                 const float* __restrict__ bt,
                                                 float* __restrict__ c) {
    __shared__ float red[256];
    const int row = blockIdx.x;
    const int t   = threadIdx.x;
    const float* xr = x + (size_t)row * DIM;
    float a = xr[t], b = xr[t + 256];

    red[t] = a + b;
    __syncthreads();
    for (int s = 128; s > 0; s >>= 1) {
        if (t < s) red[t] += red[t + s];
        __syncthreads();
    }
    const float mu = red[0] * (1.0f / DIM);
    __syncthreads();

    const float da = a - mu, db = b - mu;
    red[t] = da * da + db * db;
    __syncthreads();
    for (int s = 128; s > 0; s >>= 1) {
        if (t < s) red[t] += red[t + s];
        __syncthreads();
    }
    const float rs = rsqrtf(red[0] * (1.0f / DIM) + 1e-5f);

    float* cr = c + (size_t)row * DIM;
    cr[t]       = da * rs * g[t]       + bt[t];
    cr[t + 256] = db * rs * g[t + 256] + bt[t + 256];
}

// ---------------------------------------------------------------------------
// Y[m,n] = sum_k X[m,k] * W[n,k] + bias[n];  M=4096, K=512, Nout=512.
// One wave (32 lanes) per 16x16 output tile, WMMA f32 16x16x4 over K.
// A frag (16x4): lanes 0-15 -> M=lane, K=k0,k0+1 ; lanes 16-31 -> K=k0+2,k0+3.
// B frag (4x16): lanes 0-15 -> N=lane, K=k0,k0+1 ; lanes 16-31 -> K=k0+2,k0+3.
// Both are contiguous float2 loads (K is contiguous in X and W rows).
// ---------------------------------------------------------------------------
__global__ __launch_bounds__(256) void gemm_qkv(const float* __restrict__ X,
                                                const float* __restrict__ W,
                                                const float* __restrict__ bias,
                                                float* __restrict__ Y) {
    const int lane = threadIdx.x;          // 0..31
    const int wave = threadIdx.y;          // 0..7
    const int tile = blockIdx.x * 8 + wave;   // 0..8191
    const int nt = tile & 31;              // 32 N-tiles
    const int mt = tile >> 5;              // 256 M-tiles
    const int m0 = mt * 16, n0 = nt * 16;
    const int lm    = lane & 15;
    const int khalf = (lane >> 4) << 1;    // 0 or 2

    const float* aptr = X + (size_t)(m0 + lm) * DIM + khalf;
    const float* bptr = W + (size_t)(n0 + lm) * DIM + khalf;

    v8f acc = {};
#pragma unroll 8
    for (int k0 = 0; k0 < DIM; k0 += 4) {
        v2f a = *(const v2f*)(aptr + k0);
        v2f b = *(const v2f*)(bptr + k0);
        acc = __builtin_amdgcn_wmma_f32_16x16x4_f32(false, a, false, b,
                                                    (short)0, acc, false, false);
    }

    const float bn = bias[n0 + lm];
    const int mbase = m0 + ((lane >> 4) << 3);
    float* yp = Y + (size_t)mbase * DIM + n0 + lm;
#pragma unroll
    for (int r = 0; r < 8; ++r) yp[(size_t)r * DIM] = acc[r] + bn;
}

// ---------------------------------------------------------------------------
// Attention for one (batch, head, 16-row block of N).
// Stage 1: scores[16 x 1024] = (q @ k^T) * scale  -> LDS (8 waves x 8 col tiles)
// Stage 2: softmax over m, dist = adj * p, write dist to global (NT) + LDS
// Stage 3: att[16 x 64] = dist @ v (waves 0-3, one 16x16 tile each, K=1024)
// dist/att are write-once outputs: stream them with non-temporal stores so the
// 142 MB of output doesn't evict the L2-resident q/k/v/c working set (32 MB).
// ---------------------------------------------------------------------------
__global__ __launch_bounds__(256) void attn_kernel(const float* __restrict__ q,
                                                   const float* __restrict__ k,
                                                   const float* __restrict__ v,
                                                   const float* __restrict__ adj,
                                                   float* __restrict__ att_out,
                                                   float* __restrict__ dist_out) {
    __shared__ float sc[16 * SCPAD];   // 65792 B (<= 320 KB/WGP on CDNA5)

    const int lane = threadIdx.x;       // 0..31
    const int wave = threadIdx.y;       // 0..7
    const int nb = blockIdx.x;          // 0..63 row block
    const int h  = blockIdx.y;          // 0..7
    const int b  = blockIdx.z;          // 0..3
    const int n0 = nb * 16;
    const int lm    = lane & 15;
    const int khalf = (lane >> 4) << 1;

    // ---- Stage 1: scores -> LDS ----
    const float* qa = q + ((size_t)(b * SEQ + n0 + lm)) * DIM + h * HDK + khalf;
#pragma unroll 1
    for (int t = 0; t < 8; ++t) {
        const int m0 = (wave * 8 + t) * 16;
        const float* kb = k + ((size_t)(b * SEQ + m0 + lm)) * DIM + h * HDK + khalf;
        v8f acc = {};
#pragma unroll
        for (int j0 = 0; j0 < HDK; j0 += 4) {
            v2f a  = *(const v2f*)(qa + j0);
            v2f bb = *(const v2f*)(kb + j0);
            acc = __builtin_amdgcn_wmma_f32_16x16x4_f32(false, a, false, bb,
                                                        (short)0, acc, false, false);
        }
        const int rbase = (lane >> 4) << 3;
#pragma unroll
        for (int r = 0; r < 8; ++r)
            sc[(rbase + r) * SCPAD + m0 + lm] = acc[r] * SCALE;
    }
    __syncthreads();

    // ---- Stage 2: softmax + adj mask, write dist (non-temporal) ----
    {
        const int tid = wave * 32 + lane;
        const int row = tid >> 4;          // 0..15
        const int sub = tid & 15;
        float* srow = sc + row * SCPAD;

        float mx = -1e30f;
        for (int m = sub; m < SEQ; m += 16) mx = fmaxf(mx, srow[m]);
#pragma unroll
        for (int off = 8; off > 0; off >>= 1) mx = fmaxf(mx, __shfl_xor(mx, off, 16));

        float sum = 0.0f;
        for (int m = sub; m < SEQ; m += 16) {
            float p = __expf(srow[m] - mx);
            srow[m] = p;
            sum += p;
        }
#pragma unroll
        for (int off = 8; off > 0; off >>= 1) sum += __shfl_xor(sum, off, 16);
        const float inv = 1.0f / sum;

        const int gr = n0 + row;
        const float* arow = adj + (size_t)gr * SEQ;
        float* drow = dist_out + ((size_t)(b * NHEAD + h) * SEQ + gr) * SEQ;
        for (int m = sub; m < SEQ; m += 16) {
            float d = arow[m] * srow[m] * inv;
            srow[m] = d;
            __builtin_nontemporal_store(d, drow + m);
        }
    }
    __syncthreads();

    // ---- Stage 3: att = dist @ v ----
    if (wave < 4) {
        const int jv0 = wave * 16;
        const float* vb = v + ((size_t)b * SEQ) * DIM + h * HDK + jv0 + lm;
        v8f acc = {};
#pragma unroll 4
        for (int m0 = 0; m0 < SEQ; m0 += 4) {
            const float* sp = sc + lm * SCPAD + m0 + khalf;
            v2f a;
            a.x = sp[0];
            a.y = sp[1];
            v2f bb;
            bb.x = vb[(size_t)(m0 + khalf) * DIM];
            bb.y = vb[(size_t)(m0 + khalf + 1) * DIM];
            acc = __builtin_amdgcn_wmma_f32_16x16x4_f32(false, a, false, bb,
                                                        (short)0, acc, false, false);
        }
        const int rbase = (lane >> 4) << 3;
        float* op = att_out + ((size_t)(b * SEQ + n0 + rbase)) * DIM + h * HDK + jv0 + lm;
#pragma unroll
        for (int r = 0; r < 8; ++r)
            __builtin_nontemporal_store(acc[r], op + (size_t)r * DIM);
    }
}

// ---------------------------------------------------------------------------
extern "C" void kernel_launch(void* const* d_in, const int* in_sizes, int n_in,
                              void* d_out, int out_size, void* d_ws, size_t ws_size,
                              hipStream_t stream) {
    const float* user_exo = (const float*)d_in[0];
    const float* exo      = (const float*)d_in[1];
    const float* adj      = (const float*)d_in[2];
    const float* Wq       = (const float*)d_in[3];
    const float* bq       = (const float*)d_in[4];
    const float* Wk       = (const float*)d_in[5];
    const float* bk       = (const float*)d_in[6];
    const float* Wv       = (const float*)d_in[7];
    const float* bv       = (const float*)d_in[8];
    const float* gamma    = (const float*)d_in[9];
    const float* beta     = (const float*)d_in[10];

    const size_t rows = (size_t)BATCH * SEQ;        // 4096
    float* c = (float*)d_ws;
    float* q = c + rows * DIM;
    float* k = q + rows * DIM;
    float* v = k + rows * DIM;

    float* att  = (float*)d_out;                         // [B, N, DV]
    float* dist = att + rows * DIM;                      // [B, NH, N, N]

    ln_kernel<<<dim3(BATCH * SEQ), dim3(256), 0, stream>>>(user_exo, gamma, beta, c);

    dim3 gblk(32, 8);
    gemm_qkv<<<dim3(1024), gblk, 0, stream>>>(c,   Wq, bq, q);
    gemm_qkv<<<dim3(1024), gblk, 0, stream>>>(c,   Wk, bk, k);
    gemm_qkv<<<dim3(1024), gblk, 0, stream>>>(exo, Wv, bv, v);

    attn_kernel<<<dim3(64, NHEAD, BATCH), gblk, 0, stream>>>(q, k, v, adj, att, dist);
}